// MGQA_67654324847571
// MI455X (gfx1250) — compile-verified
//
#include <hip/hip_runtime.h>
#include <hip/hip_bf16.h>
#include <math.h>

// Problem constants (from reference)
#define Bsz   64
#define Wlen  2048
#define DIM   4096
#define NH    32
#define NKVH  8
#define HD    128
#define REP   4          // NH / NKVH
#define SCALE 0.08838834764831845f  // 1/sqrt(128)

typedef float v2f __attribute__((ext_vector_type(2)));
typedef float v8f __attribute__((ext_vector_type(8)));

// ---------------------------------------------------------------------------
// CDNA5 async global->LDS copy (16B per lane), tracked by ASYNCcnt.
// Generic shared-pointer low 32 bits == LDS byte offset (flat LDS aperture).
// ---------------------------------------------------------------------------
__device__ __forceinline__ void async_copy16(float* lds_dst, const float* gsrc)
{
    unsigned loff = (unsigned)(uintptr_t)lds_dst;
    unsigned long long ga = (unsigned long long)(uintptr_t)gsrc;
    asm volatile("global_load_async_to_lds_b128 %0, %1, off"
                 :: "v"(loff), "v"(ga) : "memory");
}

__device__ __forceinline__ void wait_async0()
{
#if __has_builtin(__builtin_amdgcn_s_wait_asynccnt)
    __builtin_amdgcn_s_wait_asynccnt(0);
#else
    asm volatile("s_wait_asynccnt 0" ::: "memory");
#endif
}

// ---------------------------------------------------------------------------
// GEMM  C[64 x N] = A[64 x K] @ B[K x N]   (fp32, WMMA 16x16x4)
// grid = N/64 blocks, block = 128 (4 waves). A block covers all 64 rows and a
// 64-wide N strip: each wave owns 16 columns across four 16-row M tiles
// (4 accumulators). The A chunk (64x64) is shared by all waves and staged
// once per block via async global->LDS copies (padded stride 68 => the
// 16-lane A-fragment reads land on distinct banks). Weights are therefore
// read exactly once grid-wide; A traffic is cut 4x vs per-wave loads.
// ---------------------------------------------------------------------------
template<int N>
__global__ __launch_bounds__(128)
void gemm64_wmma(const float* __restrict__ A, const float* __restrict__ Bm,
                 float* __restrict__ C, int K)
{
    __shared__ float at[64 * 68];

    const int tid  = threadIdx.x;
    const int lane = tid & 31;
    const int wv   = tid >> 5;
    const int n0   = blockIdx.x * 64 + wv * 16;

    const int kboff = (lane >> 4) << 1;          // lanes 16..31 hold K+2,K+3
    const int col   = n0 + (lane & 15);          // B-fragment / C column
    const int mrow  = lane & 15;

    v8f acc0 = {}, acc1 = {}, acc2 = {}, acc3 = {};

#pragma unroll 1
    for (int k0 = 0; k0 < K; k0 += 64) {
        // Stage A chunk (64 rows x 64 k) via async copies, coalesced.
        for (int i = tid; i < 64 * 16; i += 128) {
            const int e = i * 4;
            const int m  = e >> 6;      // 0..63
            const int kk = e & 63;      // 0..60 step 4
            async_copy16(&at[m * 68 + kk], A + (size_t)m * K + k0 + kk);
        }
        wait_async0();
        __syncthreads();

        const float* Bp = Bm + (size_t)(k0 + kboff) * N + col;
#pragma unroll 4
        for (int kk = 0; kk < 64; kk += 4) {
            v2f b; b.x = Bp[0]; b.y = Bp[N];
            __builtin_prefetch(Bp + (size_t)64 * N, 0, 3);   // stream-ahead
            const float* ar = &at[mrow * 68 + kk + kboff];
            v2f a0; a0.x = ar[0];           a0.y = ar[1];
            v2f a1; a1.x = ar[16 * 68];     a1.y = ar[16 * 68 + 1];
            v2f a2; a2.x = ar[32 * 68];     a2.y = ar[32 * 68 + 1];
            v2f a3; a3.x = ar[48 * 68];     a3.y = ar[48 * 68 + 1];
            acc0 = __builtin_amdgcn_wmma_f32_16x16x4_f32(false, a0, false, b,
                                                         (short)0, acc0, false, false);
            acc1 = __builtin_amdgcn_wmma_f32_16x16x4_f32(false, a1, false, b,
                                                         (short)0, acc1, false, false);
            acc2 = __builtin_amdgcn_wmma_f32_16x16x4_f32(false, a2, false, b,
                                                         (short)0, acc2, false, false);
            acc3 = __builtin_amdgcn_wmma_f32_16x16x4_f32(false, a3, false, b,
                                                         (short)0, acc3, false, false);
            Bp += (size_t)4 * N;
        }
        __syncthreads();
    }

    // D layout: VGPR j -> row j (lanes 0-15) / row 8+j (lanes 16-31)
    const int srow = (lane >> 4) << 3;
#pragma unroll
    for (int j = 0; j < 8; ++j) {
        C[(size_t)(srow + j) * N + col]        = acc0[j];
        C[(size_t)(16 + srow + j) * N + col]   = acc1[j];
        C[(size_t)(32 + srow + j) * N + col]   = acc2[j];
        C[(size_t)(48 + srow + j) * N + col]   = acc3[j];
    }
}

// ---------------------------------------------------------------------------
// RoPE (interleaved pairs): t[2i],t[2i+1] rotated by (cos,sin)[b,i]
// ---------------------------------------------------------------------------
__global__ __launch_bounds__(256)
void rope_kernel(float* __restrict__ xq, float* __restrict__ xk,
                 const float* __restrict__ cosb, const float* __restrict__ sinb)
{
    const int QP = Bsz * NH   * (HD / 2);
    const int KP = Bsz * NKVH * (HD / 2);
    int tid = blockIdx.x * blockDim.x + threadIdx.x;

    float* p;
    int b, i;
    if (tid < QP) {
        i = tid & 63;
        int h = (tid >> 6) & (NH - 1);
        b = tid / (64 * NH);
        p = xq + ((size_t)b * NH + h) * HD + 2 * i;
    } else if (tid < QP + KP) {
        int t = tid - QP;
        i = t & 63;
        int h = (t >> 6) & (NKVH - 1);
        b = t / (64 * NKVH);
        p = xk + ((size_t)b * NKVH + h) * HD + 2 * i;
    } else {
        return;
    }
    const float c = cosb[b * 64 + i];
    const float s = sinb[b * 64 + i];
    const float t0 = p[0], t1 = p[1];
    p[0] = t0 * c - t1 * s;
    p[1] = t0 * s + t1 * c;
}

// ---------------------------------------------------------------------------
// Scores: S[b,g,r,w] = (Q[b,g,r,:] . K[b,w,g,:]) * SCALE
// K row `pos` substituted with freshly projected xk (functional cache update).
// grid = (W/64, NKVH, B), block = 128.  K tile (64 x 128) staged into LDS with
// async global->LDS b128 copies; 132-float row stride keeps B-fragment reads
// bank-conflict-free.  M padded 4 -> 16 via lane&3 row replication.
// ---------------------------------------------------------------------------
__global__ __launch_bounds__(128)
void attn_scores(const float* __restrict__ xq, const float* __restrict__ xk,
                 const float* __restrict__ cache_k, const int* __restrict__ posp,
                 float* __restrict__ scores)
{
    __shared__ float kt[64 * 132];
    __shared__ float qt[4 * 132];

    const int b  = blockIdx.z;
    const int g  = blockIdx.y;
    const int w0 = blockIdx.x * 64;
    const int pos = *posp;
    const int tid = threadIdx.x;

    for (int i = tid; i < 64 * (HD / 4); i += 128) {
        const int e  = i * 4;
        const int wl = e >> 7;          // 0..63
        const int d  = e & (HD - 1);    // 0..124 step 4
        const int w  = w0 + wl;
        const float* src = (w == pos)
            ? (xk + ((size_t)b * NKVH + g) * HD + d)
            : (cache_k + (((size_t)b * Wlen + w) * NKVH + g) * HD + d);
        async_copy16(&kt[wl * 132 + d], src);
    }
    for (int i = tid; i < 4 * (HD / 4); i += 128) {
        const int e = i * 4;
        const int r = e >> 7;
        const int d = e & (HD - 1);
        async_copy16(&qt[r * 132 + d],
                     xq + ((size_t)b * NH + (g * REP + r)) * HD + d);
    }
    wait_async0();
    __syncthreads();

    const int lane  = tid & 31;
    const int wvid  = tid >> 5;
    const int wb    = wvid * 16;              // this wave's W sub-tile
    const int kboff = (lane >> 4) << 1;
    const int r     = lane & 3;               // padded M: rows replicated mod 4
    const int colw  = wb + (lane & 15);

    v8f acc = {};
#pragma unroll 4
    for (int kk = 0; kk < HD; kk += 4) {
        const int kb = kk + kboff;
        v2f a;  a.x  = qt[r * 132 + kb];     a.y  = qt[r * 132 + kb + 1];
        v2f bf; bf.x = kt[colw * 132 + kb];  bf.y = kt[colw * 132 + kb + 1];
        acc = __builtin_amdgcn_wmma_f32_16x16x4_f32(false, a, false, bf,
                                                    (short)0, acc, false, false);
    }

    if (lane < 16) {
        float* dst = scores + (((size_t)(b * NKVH + g)) * REP) * Wlen + w0 + wb + lane;
#pragma unroll
        for (int j = 0; j < REP; ++j)
            dst[(size_t)j * Wlen] = acc[j] * SCALE;
    }
}

// ---------------------------------------------------------------------------
// Row softmax over W=2048. grid = B*NH rows, block = 256 (8 elems/thread).
// ---------------------------------------------------------------------------
__global__ __launch_bounds__(256)
void softmax_w(float* __restrict__ scores)
{
    __shared__ float red[256];
    float* row = scores + (size_t)blockIdx.x * Wlen;
    const int t = threadIdx.x;

    float v[8];
    float m = -3.402823466e38f;
#pragma unroll
    for (int i = 0; i < 8; ++i) { v[i] = row[t + i * 256]; m = fmaxf(m, v[i]); }
    red[t] = m; __syncthreads();
    for (int s = 128; s > 0; s >>= 1) {
        if (t < s) red[t] = fmaxf(red[t], red[t + s]);
        __syncthreads();
    }
    m = red[0]; __syncthreads();

    float sum = 0.f;
#pragma unroll
    for (int i = 0; i < 8; ++i) { v[i] = __expf(v[i] - m); sum += v[i]; }
    red[t] = sum; __syncthreads();
    for (int s = 128; s > 0; s >>= 1) {
        if (t < s) red[t] += red[t + s];
        __syncthreads();
    }
    const float inv = 1.0f / red[0];
#pragma unroll
    for (int i = 0; i < 8; ++i) row[t + i * 256] = v[i] * inv;
}

// ---------------------------------------------------------------------------
// out[b, g*REP+r, :] = P[b,g,r,:] @ V[b,:,g,:]   (V row `pos` <- xv)
// grid = (NKVH, B), block = 128.  All 4 waves walk the same W chunks (V chunk
// shared through LDS via async copies); each wave owns 2 of the 8 dim-tiles.
// ---------------------------------------------------------------------------
__global__ __launch_bounds__(128)
void attn_pv(const float* __restrict__ scores, const float* __restrict__ xv,
             const float* __restrict__ cache_v, const int* __restrict__ posp,
             float* __restrict__ out)
{
    __shared__ float vt[64 * 132];

    const int g = blockIdx.x;
    const int b = blockIdx.y;
    const int pos = *posp;
    const int tid  = threadIdx.x;
    const int lane = tid & 31;
    const int wvid = tid >> 5;

    const int d0    = wvid * 32;            // two 16-wide dim tiles per wave
    const int kboff = (lane >> 4) << 1;
    const int c0    = d0 + (lane & 15);
    const int c1    = d0 + 16 + (lane & 15);
    const float* prow = scores + ((size_t)(b * NKVH + g) * REP + (lane & 3)) * Wlen;

    v8f acc0 = {}, acc1 = {};

#pragma unroll 1
    for (int wc = 0; wc < Wlen; wc += 64) {
        for (int i = tid; i < 64 * (HD / 4); i += 128) {
            const int e  = i * 4;
            const int wl = e >> 7;
            const int d  = e & (HD - 1);
            const int w  = wc + wl;
            const float* src = (w == pos)
                ? (xv + ((size_t)b * NKVH + g) * HD + d)
                : (cache_v + (((size_t)b * Wlen + w) * NKVH + g) * HD + d);
            async_copy16(&vt[wl * 132 + d], src);
        }
        wait_async0();
        __syncthreads();

#pragma unroll 4
        for (int kk = 0; kk < 64; kk += 4) {
            const int kb = kk + kboff;
            v2f a;  a.x  = prow[wc + kb];          a.y  = prow[wc + kb + 1];
            v2f b0; b0.x = vt[kb * 132 + c0];      b0.y = vt[(kb + 1) * 132 + c0];
            acc0 = __builtin_amdgcn_wmma_f32_16x16x4_f32(false, a, false, b0,
                                                         (short)0, acc0, false, false);
            v2f b1; b1.x = vt[kb * 132 + c1];      b1.y = vt[(kb + 1) * 132 + c1];
            acc1 = __builtin_amdgcn_wmma_f32_16x16x4_f32(false, a, false, b1,
                                                         (short)0, acc1, false, false);
        }
        __syncthreads();
    }

    if (lane < 16) {
        float* dst = out + (size_t)b * (NH * HD) + (size_t)(g * REP) * HD;
#pragma unroll
        for (int j = 0; j < REP; ++j) {
            dst[(size_t)j * HD + c0] = acc0[j];
            dst[(size_t)j * HD + c1] = acc1[j];
        }
    }
}

// ---------------------------------------------------------------------------
// Host-side launch
// ---------------------------------------------------------------------------
extern "C" void kernel_launch(void* const* d_in, const int* in_sizes, int n_in,
                              void* d_out, int out_size, void* d_ws, size_t ws_size,
                              hipStream_t stream)
{
    const float* x       = (const float*)d_in[0];
    const float* cosb    = (const float*)d_in[1];
    const float* sinb    = (const float*)d_in[2];
    const float* cache_k = (const float*)d_in[3];
    const float* cache_v = (const float*)d_in[4];
    const float* wq      = (const float*)d_in[5];
    const float* wk      = (const float*)d_in[6];
    const float* wv      = (const float*)d_in[7];
    const float* wo      = (const float*)d_in[8];
    const int*   posp    = (const int*)d_in[9];

    float* ws = (float*)d_ws;
    float* xq       = ws;                         // 64*4096
    float* xk       = xq + (size_t)Bsz * DIM;     // 64*1024
    float* xv       = xk + (size_t)Bsz * NKVH * HD;
    float* scores   = xv + (size_t)Bsz * NKVH * HD;   // 64*8*4*2048
    float* attn_out = scores + (size_t)Bsz * NKVH * REP * Wlen;  // 64*4096
    float* outp     = (float*)d_out;

    // 1) Projections (memory-bound on weights; fp32 WMMA keeps full precision)
    gemm64_wmma<DIM><<<dim3(DIM / 64), 128, 0, stream>>>(x, wq, xq, DIM);
    gemm64_wmma<NKVH * HD><<<dim3((NKVH * HD) / 64), 128, 0, stream>>>(x, wk, xk, DIM);
    gemm64_wmma<NKVH * HD><<<dim3((NKVH * HD) / 64), 128, 0, stream>>>(x, wv, xv, DIM);

    // 2) RoPE on xq, xk
    {
        const int total = Bsz * NH * (HD / 2) + Bsz * NKVH * (HD / 2);
        rope_kernel<<<(total + 255) / 256, 256, 0, stream>>>(xq, xk, cosb, sinb);
    }

    // 3) Q.K^T scores (streams the 0.54 GB K cache once — the bandwidth roofline)
    attn_scores<<<dim3(Wlen / 64, NKVH, Bsz), 128, 0, stream>>>(xq, xk, cache_k, posp, scores);

    // 4) Softmax over W
    softmax_w<<<Bsz * NH, 256, 0, stream>>>(scores);

    // 5) P.V (streams the 0.54 GB V cache once)
    attn_pv<<<dim3(NKVH, Bsz), 128, 0, stream>>>(scores, xv, cache_v, posp, attn_out);

    // 6) Output projection
    gemm64_wmma<DIM><<<dim3(DIM / 64), 128, 0, stream>>>(attn_out, wo, outp, DIM);
}